// gaussian_projection_14207751815688
// MI455X (gfx1250) — compile-verified
//
#include <hip/hip_runtime.h>
#include <math.h>

#define Bb 2
#define Hh 112
#define Ww 112
#define HW (Hh * Ww)
#define GROWS (HW + 64)      // room for 64 pad rows past count (4x-unrolled tiles)
#define SEG_T 0.7f
#define EPSf 1e-7f
#define CENTER 55.5f

typedef float v2f __attribute__((ext_vector_type(2)));
typedef float v8f __attribute__((ext_vector_type(8)));

// ---------------- phase 0: zero the per-batch counters ----------------
__global__ void init_counts(int* counts) {
    if (threadIdx.x < Bb) counts[threadIdx.x] = 0;
}

// ---------------- phase 1: select points, build K=8 point features ----
// G row layout (8 floats), arranged so each lane's WMMA B-operand slots are
// one contiguous float4:
//   [a, 2b, gb, gc,  c, ga, 0, 0]
//   lanes 0-15 need  K(0,1)=(a,2b)  and K(4,5)=(gb,gc)  -> first float4
//   lanes 16-31 need K(2,3)=(c,ga)  and K(6,7)=(0,0)    -> second float4
__global__ void build_points(const float* __restrict__ variance,
                             const float* __restrict__ center_map,
                             int* __restrict__ counts,
                             float* __restrict__ G) {
    int p = blockIdx.x * blockDim.x + threadIdx.x;
    int b = blockIdx.y;
    if (p >= HW) return;
    const float* vb = variance + (size_t)b * 3 * HW;
    float cm = center_map[(size_t)b * HW + p];
    float msk = (cm > SEG_T) ? 1.0f : 0.0f;
    float vx = vb[p] * msk;
    float vy = vb[HW + p] * msk;
    if (vx + vy == 0.0f) return;            // matches torch.nonzero(vx+vy)
    float v2 = vb[2 * HW + p];
    float sig = 1.0f / (1.0f + expf(-v2));
    float theta = 3.14f * sig;
    float s = sinf(theta), c0 = cosf(theta);
    float var_x = vx * vx + EPSf;
    float var_y = vy * vy + EPSf;
    float i2x = 0.5f / var_x, i2y = 0.5f / var_y;
    float a  = c0 * c0 * i2x + s * s * i2y;
    float bc = s * c0 * (i2y - i2x);        // b = -2sc/(4vx) + 2sc/(4vy)
    float cc = s * s * i2x + c0 * c0 * i2y;
    float b2 = 2.0f * bc;
    float pr = (float)(p / Ww) - CENTER;    // centered point coords
    float pc = (float)(p % Ww) - CENTER;
    float ga = -(2.0f * a * pr + b2 * pc);
    float gb = -(b2 * pr + 2.0f * cc * pc);
    float gc = a * pr * pr + b2 * pr * pc + cc * pc * pc;
    int idx = atomicAdd(&counts[b], 1);
    float* row = G + ((size_t)b * GROWS + idx) * 8;
    row[0] = a;  row[1] = b2; row[2] = gb; row[3] = gc;
    row[4] = cc; row[5] = ga; row[6] = 0.0f; row[7] = 0.0f;
}

// ---------------- phase 1b: pad to a 64-point boundary ----------------
// pad rows give q = 1e30 (never the min; exp(-1e30) -> 0 if no real points)
__global__ void pad_points(const int* __restrict__ counts, float* __restrict__ G) {
    int b = blockIdx.x;
    int r = counts[b] + (int)threadIdx.x;   // 64 threads
    if (r < GROWS) {
        float* row = G + ((size_t)b * GROWS + r) * 8;
        row[0] = 0.0f; row[1] = 0.0f; row[2] = 0.0f; row[3] = 1e30f;
        row[4] = 0.0f; row[5] = 0.0f; row[6] = 0.0f; row[7] = 0.0f;
    }
}

// ---------------- phase 2: WMMA GEMM (K=8 as 2x K=4) + row-min + exp ---
// One wave per 16-pixel tile. Point dimension consumed in 64-point
// super-tiles: 4 b128 loads issued together (4 loads in flight), then
// 4 x (2 chained v_wmma_f32_16x16x4_f32) + mins.
__global__ __launch_bounds__(256) void gauss_max_wmma(
        const float* __restrict__ G, const int* __restrict__ counts,
        float* __restrict__ out) {
    const int tilesPerB = HW / 16;                      // 784
    int wave = blockIdx.x * (blockDim.x >> 5) + (threadIdx.x >> 5);
    int b    = wave / tilesPerB;
    int tile = wave % tilesPerB;
    int L  = threadIdx.x & 31;
    int Mi = L & 15;
    bool lo = (L < 16);

    // A operand: pixel features (constant per wave). Lane layout per ISA:
    // lanes 0-15 K=(0,1), lanes 16-31 K=(2,3) for wmma #1; (4,5)/(6,7) for #2.
    int m = tile * 16 + Mi;
    float fi = (float)(m / Ww) - CENTER;
    float fj = (float)(m % Ww) - CENTER;
    v2f A1, A2;
    A1.x = lo ? fi * fi : fj * fj;   // K0 = i'^2 | K2 = j'^2
    A1.y = lo ? fi * fj : fi;        // K1 = i'j' | K3 = i'
    A2.x = lo ? fj : 0.0f;           // K4 = j'   | K6 = 0
    A2.y = lo ? 1.0f : 0.0f;         // K5 = 1    | K7 = 0

    int cnt = counts[b];
    int npt4 = (cnt + 63) >> 6;      // 64-point super-tiles (uniform per wave)
    const float4* Gb = (const float4*)(G + (size_t)b * GROWS * 8) + (lo ? 0 : 1);

    float vmin[8];
#pragma unroll
    for (int r = 0; r < 8; ++r) vmin[r] = 3.0e38f;

    for (int pt = 0; pt < npt4; ++pt) {
        int base = pt * 64 + Mi;
        float4 gv[4];
#pragma unroll
        for (int u = 0; u < 4; ++u)          // 4 loads issued back-to-back
            gv[u] = Gb[(base + u * 16) * 2];
#pragma unroll
        for (int u = 0; u < 4; ++u) {
            v2f B1; B1.x = gv[u].x; B1.y = gv[u].y;
            v2f B2; B2.x = gv[u].z; B2.y = gv[u].w;
            v8f accz = {};
            v8f acc = __builtin_amdgcn_wmma_f32_16x16x4_f32(
                false, A1, false, B1, (short)0, accz, false, false);
            acc = __builtin_amdgcn_wmma_f32_16x16x4_f32(
                false, A2, false, B2, (short)0, acc, false, false);
#pragma unroll
            for (int r = 0; r < 8; ++r) vmin[r] = fminf(vmin[r], acc[r]);
        }
    }

    // min across N lives across the 16 lanes of each half (masks 1,2,4,8
    // stay inside a half). Rows 0-7 in lanes 0-15, rows 8-15 in lanes 16-31.
#pragma unroll
    for (int r = 0; r < 8; ++r) {
        float v = vmin[r];
        v = fminf(v, __shfl_xor(v, 1, 32));
        v = fminf(v, __shfl_xor(v, 2, 32));
        v = fminf(v, __shfl_xor(v, 4, 32));
        v = fminf(v, __shfl_xor(v, 8, 32));
        if (Mi == r) {
            int row = r + (lo ? 0 : 8);
            float g = expf(-v + EPSf);       // max exp(-q) = exp(-min q)
            float o = (g >= 0.7f) ? g : 0.0f;
            out[(size_t)b * HW + tile * 16 + row] = o;
        }
    }
}

// ---------------- tail: zero the boolean flag element(s) --------------
__global__ void write_tail(float* out, int start, int n) {
    int t = blockIdx.x * blockDim.x + threadIdx.x;
    if (t < n) out[start + t] = 0.0f;
}

extern "C" void kernel_launch(void* const* d_in, const int* in_sizes, int n_in,
                              void* d_out, int out_size, void* d_ws, size_t ws_size,
                              hipStream_t stream) {
    // inputs: 0=x (unused, dead conv branch), 1=variance, 2=center_map,
    //         3=conv_w (unused), 4=conv_b (unused)
    const float* variance = (const float*)d_in[1];
    const float* center   = (const float*)d_in[2];
    float* out = (float*)d_out;

    int*   counts = (int*)d_ws;
    float* G      = (float*)((char*)d_ws + 256);   // B * GROWS * 8 floats (~807 KB)

    init_counts<<<1, 32, 0, stream>>>(counts);

    dim3 g1((HW + 255) / 256, Bb);
    build_points<<<g1, 256, 0, stream>>>(variance, center, counts, G);

    pad_points<<<Bb, 64, 0, stream>>>(counts, G);

    int waves  = Bb * (HW / 16);   // 1568 pixel tiles, one wave each
    int blocks = waves / 8;        // 8 waves (256 threads) per block
    gauss_max_wmma<<<blocks, 256, 0, stream>>>(G, counts, out);

    int tail = out_size - Bb * HW; // boolean `flag` output, always False
    if (tail > 0)
        write_tail<<<(tail + 255) / 256, 256, 0, stream>>>(out, Bb * HW, tail);
}